// SymInfoNCELoss_22076131901967
// MI455X (gfx1250) — compile-verified
//
#include <hip/hip_runtime.h>
#include <math.h>

// Symmetric InfoNCE over 3 views, fused GEMM + online softmax reductions.
// Uses V_WMMA_F32_16X16X4_F32 (CDNA5 fp32 matrix path) — exact fp32 GEMM.
// All streaming loads are forced to the GLOBAL address space (addrspace(1))
// so they lower to global_load_* (LOADcnt only) instead of flat_load_*
// (which would tie up LOADcnt+DScnt and the LDS arbitration path).

typedef float v2f __attribute__((ext_vector_type(2)));
typedef float v4f __attribute__((ext_vector_type(4)));   // native vec, not HIP class
typedef float v8f __attribute__((ext_vector_type(8)));

#define GAS __attribute__((address_space(1)))
typedef const float GAS* gcfloat;
typedef const v2f   GAS* gcv2f;
typedef const v4f   GAS* gcv4f;

#define BDIM 4096
#define DDIM 512
#define NPAIR 3

// ---------------------------------------------------------------------------
// Kernel 1: per-row inverse norms for all 3 views: inv[v*B + b] = 1/max(||z||,eps)
// One wave per row; coalesced global b128 loads; shfl-xor wave reduction.
// ---------------------------------------------------------------------------
__global__ __launch_bounds__(256) void simloss_norm_kernel(
    const float* __restrict__ z0, const float* __restrict__ z1,
    const float* __restrict__ z2, float* __restrict__ inv) {
  const int wid  = threadIdx.x >> 5;
  const int lane = threadIdx.x & 31;
  const int row  = blockIdx.x * 8 + wid;            // 0 .. 3*4096-1
  const int v    = row / BDIM;
  const float* srcg = (v == 0) ? z0 : ((v == 1) ? z1 : z2);
  gcfloat src = (gcfloat)(srcg + (size_t)(row % BDIM) * DDIM);

  float ss = 0.f;
#pragma unroll
  for (int t = 0; t < 4; ++t) {
    const v4f x = *(gcv4f)(src + lane * 4 + t * 128);
    ss = fmaf(x.x, x.x, fmaf(x.y, x.y, fmaf(x.z, x.z, fmaf(x.w, x.w, ss))));
  }
  ss += __shfl_xor(ss, 1, 32);
  ss += __shfl_xor(ss, 2, 32);
  ss += __shfl_xor(ss, 4, 32);
  ss += __shfl_xor(ss, 8, 32);
  ss += __shfl_xor(ss, 16, 32);
  if (lane == 0) inv[row] = 1.f / fmaxf(sqrtf(ss), 1e-8f);
}

// ---------------------------------------------------------------------------
// Kernel 2: zero the row/col exp-sum accumulators and the output scalar.
// Must run every launch (graph replays don't re-poison, and we accumulate
// with atomics).
// ---------------------------------------------------------------------------
__global__ void simloss_zero_kernel(float* __restrict__ sums,
                                    float* __restrict__ out, int n) {
  const int idx = blockIdx.x * blockDim.x + threadIdx.x;
  if (idx < n) sums[idx] = 0.f;
  if (idx == 0) out[0] = 0.f;
}

// ---------------------------------------------------------------------------
// Kernel 3: fused GEMM (fp32 WMMA) + exp row/col sum accumulation.
// Block = 128 threads (4 waves). Wave tile = 64x64 (4x4 of 16x16 WMMA tiles).
// Block tile = 128x128. Grid = (32, 32, 3 pairs).
// sim[b,c] = 10 * dot(Zi[b], Zj[c]) * inv_i[b] * inv_j[c]; we accumulate
// exp(sim - 10) per row and per column (logits are <= 10, so this is a safe
// fixed-shift logsumexp: lse = log(sum) + 10).
// ---------------------------------------------------------------------------
__global__ __launch_bounds__(128) void simloss_gemm_kernel(
    const float* __restrict__ z0, const float* __restrict__ z1,
    const float* __restrict__ z2, const float* __restrict__ inv,
    float* __restrict__ rs, float* __restrict__ cs) {
  const int p  = blockIdx.z;                // pair id: (0,1),(0,2),(1,2)
  const int vi = (p == 2) ? 1 : 0;
  const int vj = (p == 0) ? 1 : 2;
  const float* Zi = (vi == 0) ? z0 : z1;
  const float* Zj = (vj == 1) ? z1 : z2;
  gcfloat inv_i = (gcfloat)(inv + vi * BDIM);
  gcfloat inv_j = (gcfloat)(inv + vj * BDIM);
  float* rowSum = rs + p * BDIM;
  float* colSum = cs + p * BDIM;

  const int lane = threadIdx.x & 31;
  const int w    = threadIdx.x >> 5;
  const int wm   = w >> 1;
  const int wn   = w & 1;
  const int rowBase = blockIdx.y * 128 + wm * 64;
  const int colBase = blockIdx.x * 128 + wn * 64;
  const int r16  = lane & 15;
  const int lh   = lane >> 4;               // lane half: 0 or 1
  const int koff = lh * 2;                  // fp32 A/B frag: half 0 -> K{0,1}, half 1 -> K{2,3}

  gcfloat Abase = (gcfloat)(Zi + (size_t)(rowBase + r16) * DDIM + koff);
  gcfloat Bbase = (gcfloat)(Zj + (size_t)(colBase + r16) * DDIM + koff);

  v8f acc[4][4];
  const v8f vzero = {0.f, 0.f, 0.f, 0.f, 0.f, 0.f, 0.f, 0.f};
#pragma unroll
  for (int m = 0; m < 4; ++m)
#pragma unroll
    for (int n = 0; n < 4; ++n) acc[m][n] = vzero;

  for (int k = 0; k < DDIM; k += 4) {
    v2f a[4], b[4];
#pragma unroll
    for (int m = 0; m < 4; ++m)
      a[m] = *(gcv2f)(Abase + (size_t)m * 16 * DDIM + k);
#pragma unroll
    for (int n = 0; n < 4; ++n)
      b[n] = *(gcv2f)(Bbase + (size_t)n * 16 * DDIM + k);
#pragma unroll
    for (int m = 0; m < 4; ++m)
#pragma unroll
      for (int n = 0; n < 4; ++n)
        acc[m][n] = __builtin_amdgcn_wmma_f32_16x16x4_f32(
            false, a[m], false, b[n], (short)0, acc[m][n], false, false);
  }

  // ---- epilogue: scale, exp, row/col reductions ----
  float cj[4];
#pragma unroll
  for (int n = 0; n < 4; ++n) cj[n] = inv_j[colBase + n * 16 + r16];

  float colAcc[4] = {0.f, 0.f, 0.f, 0.f};

#pragma unroll
  for (int m = 0; m < 4; ++m) {
    float ii[8];
#pragma unroll
    for (int r = 0; r < 8; ++r)
      ii[r] = inv_i[rowBase + m * 16 + r + 8 * lh];

    float er[8] = {0.f, 0.f, 0.f, 0.f, 0.f, 0.f, 0.f, 0.f};
#pragma unroll
    for (int n = 0; n < 4; ++n) {
#pragma unroll
      for (int r = 0; r < 8; ++r) {
        // C/D layout: VGPR r, lanes 0-15 -> row m*16+r,   col n*16+lane
        //                     lanes 16-31 -> row m*16+8+r, col n*16+lane-16
        const float e = __expf(fmaf(10.f * acc[m][n][r], ii[r] * cj[n], -10.f));
        er[r] += e;
        colAcc[n] += e;
      }
    }
    // Reduce row partials across the 16 lanes of each half (columns of tile).
#pragma unroll
    for (int r = 0; r < 8; ++r) {
      float x = er[r];
      x += __shfl_xor(x, 1, 32);
      x += __shfl_xor(x, 2, 32);
      x += __shfl_xor(x, 4, 32);
      x += __shfl_xor(x, 8, 32);
      if (r16 == 0)
        atomicAdd(&rowSum[rowBase + m * 16 + r + 8 * lh], x);
    }
  }
  // Column partials: combine lane halves (same column set), 16 lanes emit.
#pragma unroll
  for (int n = 0; n < 4; ++n) {
    float x = colAcc[n];
    x += __shfl_xor(x, 16, 32);
    if (lh == 0) atomicAdd(&colSum[colBase + n * 16 + r16], x);
  }
}

// ---------------------------------------------------------------------------
// Kernel 4: final per-row loss terms + reduction to scalar.
// loss = (1/3) * mean_b over pairs of [ 0.5*(log rs + log cs) + 10 - diag ]
// where diag = 10 * dot(z_i[b], z_j[b]) * inv_i[b] * inv_j[b].
// One wave per row b.
// ---------------------------------------------------------------------------
__global__ __launch_bounds__(256) void simloss_final_kernel(
    const float* __restrict__ z0, const float* __restrict__ z1,
    const float* __restrict__ z2, const float* __restrict__ inv,
    const float* __restrict__ rs, const float* __restrict__ cs,
    float* __restrict__ out) {
  const int wid  = threadIdx.x >> 5;
  const int lane = threadIdx.x & 31;
  const int b    = blockIdx.x * 8 + wid;

  float local = 0.f;
#pragma unroll
  for (int p = 0; p < NPAIR; ++p) {
    const int vi = (p == 2) ? 1 : 0;
    const int vj = (p == 0) ? 1 : 2;
    const float* ag = (vi == 0) ? z0 : z1;
    const float* cg = (vj == 1) ? z1 : z2;
    gcfloat a = (gcfloat)(ag + (size_t)b * DDIM);
    gcfloat c = (gcfloat)(cg + (size_t)b * DDIM);
    float dot = 0.f;
#pragma unroll
    for (int t = 0; t < 4; ++t) {
      const v4f av = *(gcv4f)(a + lane * 4 + t * 128);
      const v4f cv = *(gcv4f)(c + lane * 4 + t * 128);
      dot = fmaf(av.x, cv.x, fmaf(av.y, cv.y, fmaf(av.z, cv.z, fmaf(av.w, cv.w, dot))));
    }
    dot += __shfl_xor(dot, 1, 32);
    dot += __shfl_xor(dot, 2, 32);
    dot += __shfl_xor(dot, 4, 32);
    dot += __shfl_xor(dot, 8, 32);
    dot += __shfl_xor(dot, 16, 32);
    if (lane == 0) {
      const float diag = 10.f * dot * inv[vi * BDIM + b] * inv[vj * BDIM + b];
      local += 0.5f * (logf(rs[p * BDIM + b]) + logf(cs[p * BDIM + b])) + 10.f - diag;
    }
  }
  if (lane == 0) atomicAdd(out, local * (1.f / (3.f * (float)BDIM)));
}

// ---------------------------------------------------------------------------
extern "C" void kernel_launch(void* const* d_in, const int* in_sizes, int n_in,
                              void* d_out, int out_size, void* d_ws, size_t ws_size,
                              hipStream_t stream) {
  (void)in_sizes; (void)n_in; (void)out_size; (void)ws_size;
  const float* z0 = (const float*)d_in[0];
  const float* z1 = (const float*)d_in[1];
  const float* z2 = (const float*)d_in[2];
  float* out = (float*)d_out;

  // Workspace layout (floats): inv[3*4096] | rs[3*4096] | cs[3*4096]  (~144 KB)
  float* W   = (float*)d_ws;
  float* inv = W;
  float* rsv = W + 3 * BDIM;
  float* csv = W + 6 * BDIM;

  simloss_norm_kernel<<<dim3(3 * BDIM / 8), 256, 0, stream>>>(z0, z1, z2, inv);
  simloss_zero_kernel<<<dim3((2 * 3 * BDIM + 255) / 256), 256, 0, stream>>>(
      rsv, out, 2 * 3 * BDIM);
  simloss_gemm_kernel<<<dim3(BDIM / 128, BDIM / 128, NPAIR), 128, 0, stream>>>(
      z0, z1, z2, inv, rsv, csv);
  simloss_final_kernel<<<dim3(BDIM / 8), 256, 0, stream>>>(
      z0, z1, z2, inv, rsv, csv, out);
}